// Attention_25082609009272
// MI455X (gfx1250) — compile-verified
//
#include <hip/hip_runtime.h>

typedef float v2f __attribute__((ext_vector_type(2)));
typedef float v8f __attribute__((ext_vector_type(8)));

#define BATCH 16
#define SEQ   1024
#define CH    256
#define QS    258    // LDS row stride (floats) for Q tile: 258 mod 64 = 2 -> conflict-free column reads
#define SS    1026   // LDS row stride (floats) for score tile: same trick, keeps 8B alignment

__global__ __launch_bounds__(256)
void attn_fused_f32wmma(const float* __restrict__ query,
                        const float* __restrict__ key,
                        const float* __restrict__ value,
                        const unsigned char* __restrict__ mask,  // jnp.bool_ -> 1 byte
                        const float* __restrict__ dis,
                        float* __restrict__ p_val,
                        float* __restrict__ p_attn)
{
    __shared__ float sQ[16 * QS];   // Q tile  16 x 256
    __shared__ float sS[16 * SS];   // scores / probabilities 16 x 1024

    const int b      = blockIdx.x >> 6;          // batch
    const int q0     = (blockIdx.x & 63) << 4;   // query tile base
    const int tid    = threadIdx.x;
    const int lane   = tid & 31;
    const int wave   = tid >> 5;                 // 8 waves
    const int lane16 = lane & 15;
    const int half   = lane >> 4;                // 0: lanes 0-15, 1: lanes 16-31
    const int ks     = half * 2;                 // K-subindex base for A/B fragments

    // ---------- stage Q tile into LDS ----------
    {
        const float* qg = query + ((size_t)b * SEQ + q0) * CH;
        for (int i = tid; i < 16 * 128; i += 256) {
            const int r = i >> 7;
            const int c = (i & 127) * 2;
            sQ[r * QS + c]     = qg[r * CH + c];
            sQ[r * QS + c + 1] = qg[r * CH + c + 1];
        }
    }
    __syncthreads();

    // ---------- phase 1: scores = mask((Q K^T + dis) / 16) ----------
    // Each wave owns key-tiles kt = wave, wave+8, ...
    for (int kt = wave; kt < 64; kt += 8) {
        const int k0 = kt << 4;
        // B fragment source: B[k][n] = K[k0+n][c0+k]; lane16 = n, ks selects K pair
        const float* krow = key + ((size_t)b * SEQ + k0 + lane16) * CH + ks;
        // A fragment source: A[m][k] = Q[q0+m][c0+k]; lane16 = m
        const float* arow = &sQ[lane16 * QS + ks];

        v8f acc = {};
        #pragma unroll 8
        for (int c0 = 0; c0 < CH; c0 += 4) {
            v2f a, bm;
            a.x  = arow[c0];
            a.y  = arow[c0 + 1];
            bm.x = krow[c0];
            bm.y = krow[c0 + 1];
            acc = __builtin_amdgcn_wmma_f32_16x16x4_f32(false, a, false, bm,
                                                        (short)0, acc, false, false);
        }

        // C/D layout: VGPR v, lanes 0-15 -> M=v, lanes 16-31 -> M=v+8; N = lane16
        const int mb = half * 8;
        #pragma unroll
        for (int v = 0; v < 8; ++v) {
            const int M = v + mb;
            const size_t gidx = ((size_t)b * SEQ + q0 + M) * SEQ + k0 + lane16;
            // dis/mask are streamed exactly once -> non-temporal, keep K/V in cache
            float s = (acc[v] + __builtin_nontemporal_load(&dis[gidx])) * 0.0625f;
            if (__builtin_nontemporal_load(&mask[gidx])) s = -__builtin_inff();
            sS[M * SS + k0 + lane16] = s;
        }
    }
    __syncthreads();

    // ---------- phase 2: row softmax; emit p_attn; keep P in LDS ----------
    for (int r = wave * 2; r < wave * 2 + 2; ++r) {
        float vals[32];
        float m = -__builtin_inff();
        #pragma unroll
        for (int j = 0; j < 32; ++j) {
            vals[j] = sS[r * SS + lane + 32 * j];
            m = fmaxf(m, vals[j]);
        }
        #pragma unroll
        for (int off = 16; off >= 1; off >>= 1)
            m = fmaxf(m, __shfl_xor(m, off, 32));

        float sum = 0.f;
        #pragma unroll
        for (int j = 0; j < 32; ++j) {
            const float e = __expf(vals[j] - m);   // -inf -> 0
            vals[j] = e;
            sum += e;
        }
        #pragma unroll
        for (int off = 16; off >= 1; off >>= 1)
            sum += __shfl_xor(sum, off, 32);

        const float inv = 1.f / sum;
        float* orow = p_attn + ((size_t)b * SEQ + q0 + r) * SEQ;
        #pragma unroll
        for (int j = 0; j < 32; ++j) {
            const float p = vals[j] * inv;
            sS[r * SS + lane + 32 * j] = p;
            __builtin_nontemporal_store(p, &orow[lane + 32 * j]);  // write-once stream
        }
    }
    __syncthreads();

    // ---------- phase 3: p_val = P @ V ----------
    #pragma unroll
    for (int t = 0; t < 2; ++t) {
        const int c0 = wave * 32 + t * 16;
        // B fragment: B[k][n] = V[k0+k][c0+n]; lane16 = n
        const float* vcol = value + (size_t)b * SEQ * CH + c0 + lane16;
        // A fragment: A[m][k] = P[m][k0+k] from LDS
        const float* arow = &sS[lane16 * SS + ks];

        v8f acc = {};
        #pragma unroll 8
        for (int k0 = 0; k0 < SEQ; k0 += 4) {
            v2f a, bm;
            a.x  = arow[k0];
            a.y  = arow[k0 + 1];
            bm.x = vcol[(size_t)(k0 + ks) * CH];
            bm.y = vcol[(size_t)(k0 + ks + 1) * CH];
            acc = __builtin_amdgcn_wmma_f32_16x16x4_f32(false, a, false, bm,
                                                        (short)0, acc, false, false);
        }

        const int mb = half * 8;
        #pragma unroll
        for (int v = 0; v < 8; ++v) {
            const int M = v + mb;
            __builtin_nontemporal_store(
                acc[v], &p_val[((size_t)b * SEQ + q0 + M) * CH + c0 + lane16]);
        }
    }
}

extern "C" void kernel_launch(void* const* d_in, const int* in_sizes, int n_in,
                              void* d_out, int out_size, void* d_ws, size_t ws_size,
                              hipStream_t stream) {
    const float*         q    = (const float*)d_in[0];
    const float*         k    = (const float*)d_in[1];
    const float*         v    = (const float*)d_in[2];
    const unsigned char* mask = (const unsigned char*)d_in[3];
    const float*         dis  = (const float*)d_in[4];

    float* p_val  = (float*)d_out;                                  // B*N*C floats
    float* p_attn = p_val + (size_t)BATCH * SEQ * CH;               // then B*N*N floats

    dim3 grid(BATCH * (SEQ / 16));   // 1024 workgroups
    attn_fused_f32wmma<<<grid, 256, 0, stream>>>(q, k, v, mask, dis, p_val, p_attn);
}